// AbstractGenerativeUpsample_84439057039838
// MI455X (gfx1250) — compile-verified
//
#include <hip/hip_runtime.h>
#include <hip/hip_bf16.h>

typedef __attribute__((ext_vector_type(2))) float v2f;
typedef __attribute__((ext_vector_type(8))) float v8f;

#define NPTS 1048576
#define CDIM 64
#define MT   600000

__device__ __forceinline__ unsigned f2key(float f) {
    unsigned u = __float_as_uint(f);
    return (u & 0x80000000u) ? ~u : (u | 0x80000000u);
}

// ---------------- small utility kernels -------------------------------------

__global__ void zero_f32_kernel(float* __restrict__ p, int n) {
    int i = blockIdx.x * blockDim.x + threadIdx.x;
    if (i < n) __builtin_nontemporal_store(0.0f, &p[i]);
}

__global__ void scatter_target_kernel(const int* __restrict__ idx,
                                      float* __restrict__ loss_t, int m) {
    int i = blockIdx.x * blockDim.x + threadIdx.x;
    if (i < m) loss_t[idx[i]] = 1.0f;   // idempotent, no atomic needed
}

// state layout in ws (u32): [0]=prefix_bits [1]=himask [2]=k_rem [3]=shift
//                           [4]=final key K   [8..263]=histogram bins
__global__ void select_init_kernel(unsigned* __restrict__ st,
                                   const int* __restrict__ tgt_num) {
    int t = threadIdx.x;
    if (t == 0) {
        st[0] = 0u; st[1] = 0u;
        st[2] = (unsigned)(NPTS - tgt_num[0] - 1);  // 0-based order statistic
        st[3] = 24u; st[4] = 0u;
    }
    if (t < 256) st[8 + t] = 0u;
}

__global__ __launch_bounds__(256) void select_hist_kernel(
        const float* __restrict__ pred, unsigned* __restrict__ st) {
    __shared__ unsigned lb[256];
    int t = threadIdx.x;
    lb[t] = 0u;
    __syncthreads();
    unsigned prefix = st[0], himask = st[1], shift = st[3];
    int i = blockIdx.x * 256 + t;
    unsigned key = f2key(pred[i]);
    if ((key & himask) == prefix)
        atomicAdd(&lb[(key >> shift) & 0xFFu], 1u);
    __syncthreads();
    if (lb[t]) atomicAdd(&st[8 + t], lb[t]);
}

__global__ void select_scan_kernel(unsigned* __restrict__ st) {
    if (threadIdx.x == 0) {
        unsigned shift = st[3];
        unsigned krem  = st[2];
        unsigned cum = 0u, sel = 255u;
        for (int b = 0; b < 256; ++b) {
            unsigned c = st[8 + b];
            if (cum + c > krem) { sel = (unsigned)b; break; }
            cum += c;
        }
        unsigned prefix = st[0] | (sel << shift);
        st[0] = prefix;
        st[1] = st[1] | (0xFFu << shift);
        st[2] = krem - cum;
        if (shift == 0u) st[4] = prefix;     // full 32-bit key of k-th smallest
        st[3] = shift - 8u;
        for (int b = 0; b < 256; ++b) st[8 + b] = 0u;  // reset bins
    }
}

// ---------------- WMMA GEMM pass 1: pred only -------------------------------
// block = 256 thr = 8 waves; wave does 16 rows x 64 cols via V_WMMA_F32_16X16X4_F32.
// Regular-temporal fea loads: let lines linger in the 192MB L2 for pass 2 reuse.

__global__ __launch_bounds__(256) void gemm_pred_kernel(
        const float* __restrict__ fea, const float* __restrict__ W_up,
        const float* __restrict__ b_up, const float* __restrict__ W_cls,
        const float* __restrict__ b_cls, float* __restrict__ pred) {
    __shared__ float Wl[CDIM * CDIM];            // 16 KB, shared by 8 waves
    const int tid = threadIdx.x;
#pragma unroll
    for (int i = 0; i < 16; ++i) Wl[tid + i * 256] = W_up[tid + i * 256];
    __syncthreads();

    const int wave = tid >> 5;
    const int lane = tid & 31;
    const int half = lane >> 4;                  // lanes 16-31 hold K+2 / M+8
    const int l16  = lane & 15;
    const long long m0 = (long long)blockIdx.x * 128 + wave * 16;

    // A fragments: lane(l16) = row, half selects K pair within each 4-wide step
    const float* arow = fea + (m0 + l16) * CDIM;
    v2f a[16];
#pragma unroll
    for (int ks = 0; ks < 16; ++ks)
        a[ks] = *reinterpret_cast<const v2f*>(arow + ks * 4 + half * 2);

    float s[8];
#pragma unroll
    for (int r = 0; r < 8; ++r) s[r] = 0.0f;

#pragma unroll
    for (int nt = 0; nt < 4; ++nt) {
        const int ncol = nt * 16 + l16;
        const float bv = b_up[ncol];
        v8f c;
#pragma unroll
        for (int i = 0; i < 8; ++i) c[i] = bv;   // bias pre-loaded into C
#pragma unroll
        for (int ks = 0; ks < 16; ++ks) {
            const int krow = ks * 4 + half * 2;
            v2f b = v2f{Wl[krow * CDIM + ncol], Wl[(krow + 1) * CDIM + ncol]};
            c = __builtin_amdgcn_wmma_f32_16x16x4_f32(
                    false, a[ks], false, b, (short)0, c, false, false);
        }
        const float wc = W_cls[ncol];
#pragma unroll
        for (int r = 0; r < 8; ++r) {
            float v = c[r] > 0.0f ? c[r] : 0.0f; // relu
            s[r] += v * wc;                       // GEMV partial for pred
        }
    }
    // butterfly reduce across each 16-lane half (wave32: masks 1,2,4,8)
    const float bc = b_cls[0];
#pragma unroll
    for (int r = 0; r < 8; ++r) {
        float v = s[r];
        v += __shfl_xor(v, 1, 32);
        v += __shfl_xor(v, 2, 32);
        v += __shfl_xor(v, 4, 32);
        v += __shfl_xor(v, 8, 32);
        s[r] = v + bc;
    }
    if (l16 == 0) {
#pragma unroll
        for (int r = 0; r < 8; ++r)
            pred[m0 + half * 8 + r] = s[r];      // lane0 rows 0-7, lane16 rows 8-15
    }
}

// ---------------- WMMA GEMM pass 2: recompute fea_up, mask, write outputs ---
// fea loads: non-temporal (last use).  fea_out stores: non-temporal streaming
// writes (never re-read) so they don't evict fea/pred from the 192MB L2.

__global__ __launch_bounds__(256) void gemm_out_kernel(
        const float* __restrict__ fea, const float* __restrict__ W_up,
        const float* __restrict__ b_up, const float* __restrict__ pred,
        const float* __restrict__ loss_t, const unsigned* __restrict__ st,
        float* __restrict__ fea_out, float* __restrict__ keep_out) {
    __shared__ float Wl[CDIM * CDIM];
    const int tid = threadIdx.x;
#pragma unroll
    for (int i = 0; i < 16; ++i) Wl[tid + i * 256] = W_up[tid + i * 256];
    __syncthreads();

    const int wave = tid >> 5;
    const int lane = tid & 31;
    const int half = lane >> 4;
    const int l16  = lane & 15;
    const long long m0 = (long long)blockIdx.x * 128 + wave * 16;

    const unsigned K = st[4];                    // k-th smallest key
    float kf[8];
#pragma unroll
    for (int r = 0; r < 8; ++r) {
        const long long m = m0 + half * 8 + r;   // row owned by this lane for VGPR r
        const bool kp = (f2key(pred[m]) > K) || (loss_t[m] != 0.0f);
        kf[r] = kp ? 1.0f : 0.0f;
    }
    if (l16 == 0) {
#pragma unroll
        for (int r = 0; r < 8; ++r)
            __builtin_nontemporal_store(kf[r], &keep_out[m0 + half * 8 + r]);
    }

    const float* arow = fea + (m0 + l16) * CDIM;
    v2f a[16];
#pragma unroll
    for (int ks = 0; ks < 16; ++ks) {
        const float* p = arow + ks * 4 + half * 2;
        a[ks] = v2f{__builtin_nontemporal_load(p),
                    __builtin_nontemporal_load(p + 1)};   // last use of fea
    }

#pragma unroll
    for (int nt = 0; nt < 4; ++nt) {
        const int ncol = nt * 16 + l16;
        const float bv = b_up[ncol];
        v8f c;
#pragma unroll
        for (int i = 0; i < 8; ++i) c[i] = bv;
#pragma unroll
        for (int ks = 0; ks < 16; ++ks) {
            const int krow = ks * 4 + half * 2;
            v2f b = v2f{Wl[krow * CDIM + ncol], Wl[(krow + 1) * CDIM + ncol]};
            c = __builtin_amdgcn_wmma_f32_16x16x4_f32(
                    false, a[ks], false, b, (short)0, c, false, false);
        }
#pragma unroll
        for (int r = 0; r < 8; ++r) {
            float v = c[r] > 0.0f ? c[r] : 0.0f;          // relu
            __builtin_nontemporal_store(
                v * kf[r], &fea_out[(m0 + half * 8 + r) * CDIM + ncol]);
        }
    }
}

// ---------------- launcher ---------------------------------------------------

extern "C" void kernel_launch(void* const* d_in, const int* in_sizes, int n_in,
                              void* d_out, int out_size, void* d_ws, size_t ws_size,
                              hipStream_t stream) {
    const float* fea   = (const float*)d_in[0];
    const float* W_up  = (const float*)d_in[1];
    const float* b_up  = (const float*)d_in[2];
    const float* W_cls = (const float*)d_in[3];
    const float* b_cls = (const float*)d_in[4];
    const int*   tidx  = (const int*)d_in[5];
    const int*   tnum  = (const int*)d_in[6];

    float* out      = (float*)d_out;
    float* fea_out  = out;                                   // N*64
    float* pred     = out + (size_t)NPTS * CDIM;             // N
    float* loss_t   = pred + NPTS;                           // N
    float* keep_out = loss_t + NPTS;                         // N

    unsigned* st = (unsigned*)d_ws;

    // 1) target mask: zero + scatter
    zero_f32_kernel<<<NPTS / 256, 256, 0, stream>>>(loss_t, NPTS);
    scatter_target_kernel<<<(MT + 255) / 256, 256, 0, stream>>>(tidx, loss_t, MT);

    // 2) GEMM pass 1 -> pred
    gemm_pred_kernel<<<NPTS / 128, 256, 0, stream>>>(fea, W_up, b_up, W_cls, b_cls, pred);

    // 3) radix select of the (N - tgt - 1)-th smallest pred (exact order statistic)
    select_init_kernel<<<1, 256, 0, stream>>>(st, tnum);
    for (int p = 0; p < 4; ++p) {
        select_hist_kernel<<<NPTS / 256, 256, 0, stream>>>(pred, st);
        select_scan_kernel<<<1, 32, 0, stream>>>(st);
    }

    // 4) GEMM pass 2 -> fea_out, keep (recompute fea_up instead of spilling 268MB)
    gemm_out_kernel<<<NPTS / 128, 256, 0, stream>>>(fea, W_up, b_up, pred, loss_t,
                                                    st, fea_out, keep_out);
}